// Embedding_60327110640045
// MI455X (gfx1250) — compile-verified
//
#include <hip/hip_runtime.h>
#include <hip/hip_bf16.h>

// Embedding gather: out[t, 0:512] = weight[idx[t], 0:512]
// tokens = 8*4096 = 32768, EMBED = 512 (f32), weight = 50257 x 512 f32.
//
// Pure bandwidth kernel. One wave32 owns one token row per loop iteration:
// each lane moves 4x 16B vectors at stride 32 -> 4 fully coalesced
// 512B-per-wave b128 loads + 4 b128 non-temporal stores per row.

typedef float v4f __attribute__((ext_vector_type(4)));  // native clang vector

#define EMBED 512
#define VEC_PER_ROW (EMBED / 4)          // 128 x 16B per row
#define VEC_PER_LANE (VEC_PER_ROW / 32)  // 4 x 16B per lane

__global__ __launch_bounds__(256) void embedding_gather_kernel(
    const int* __restrict__ idx,
    const float* __restrict__ weight,
    float* __restrict__ out,
    int n_tokens)
{
  const int gtid  = blockIdx.x * blockDim.x + threadIdx.x;
  const int wave  = gtid >> 5;                       // global wave id
  const int lane  = threadIdx.x & 31;
  const int nwave = (gridDim.x * blockDim.x) >> 5;   // total waves in grid

  for (int t = wave; t < n_tokens; t += nwave) {
    // Row index is wave-uniform: force it scalar so the weight-row base
    // lands in SGPRs (saddr addressing) and only the lane offset is vector.
    const int row = __builtin_amdgcn_readfirstlane(idx[t]);

    const v4f* __restrict__ src = (const v4f*)(weight + (size_t)row * EMBED);
    v4f* __restrict__ dst       = (v4f*)(out + (size_t)t * EMBED);

    // 4 coalesced 16B loads per lane (temporal: keep weight hot in L2),
    // 4 coalesced 16B non-temporal stores (don't evict weight from L2).
    v4f v0 = src[lane + 0 * 32];
    v4f v1 = src[lane + 1 * 32];
    v4f v2 = src[lane + 2 * 32];
    v4f v3 = src[lane + 3 * 32];

    __builtin_nontemporal_store(v0, &dst[lane + 0 * 32]);
    __builtin_nontemporal_store(v1, &dst[lane + 1 * 32]);
    __builtin_nontemporal_store(v2, &dst[lane + 2 * 32]);
    __builtin_nontemporal_store(v3, &dst[lane + 3 * 32]);
  }
}

extern "C" void kernel_launch(void* const* d_in, const int* in_sizes, int n_in,
                              void* d_out, int out_size, void* d_ws, size_t ws_size,
                              hipStream_t stream) {
  const int*   idx    = (const int*)d_in[0];     // int32 token ids, 8*4096
  const float* weight = (const float*)d_in[1];   // f32, 50257*512
  float*       out    = (float*)d_out;           // f32, 32768*512

  const int n_tokens = in_sizes[0];              // 32768

  const int block = 256;                         // 8 waves per block
  const int waves_per_block = block / 32;
  int grid = (n_tokens + waves_per_block - 1) / waves_per_block;  // 4096 blocks
  if (grid < 1) grid = 1;

  embedding_gather_kernel<<<grid, block, 0, stream>>>(idx, weight, out, n_tokens);
}